// UniformLaplacianSmoothnessLoss_26628797235300
// MI455X (gfx1250) — compile-verified
//
#include <hip/hip_runtime.h>

// Problem constants from the reference (B fixed; N, F re-derived from sizes).
#define BATCH 16
#define BLK   256

// ---------------------------------------------------------------------------
// Feature detection: gfx1250 async global->LDS copy builtins
// ---------------------------------------------------------------------------
#if defined(__has_builtin)
#  if __has_builtin(__builtin_amdgcn_global_load_async_to_lds_b32)
#    define USE_ASYNC_LDS 1
#  endif
#endif
#ifndef USE_ASYNC_LDS
#  define USE_ASYNC_LDS 0
#endif

// Address-space typed pointers for the async copy builtin.
typedef __attribute__((address_space(1))) int global_int;
typedef __attribute__((address_space(3))) int lds_int;

__device__ __forceinline__ void async_wait_all() {
#if defined(__has_builtin) && __has_builtin(__builtin_amdgcn_s_wait_asynccnt)
    __builtin_amdgcn_s_wait_asynccnt(0);
#else
    asm volatile("s_wait_asynccnt 0" ::: "memory");
#endif
}

// Native L2 f32 atomic add (global_atomic_add_f32), no CAS loop.
__device__ __forceinline__ void atomAddF32(float* p, float v) {
    unsafeAtomicAdd(p, v);
}

// ---------------------------------------------------------------------------
// Kernel A: scatter neighbor sums. grid.x tiles faces in blocks of BLK,
// grid.y = batch. Face indices for the block are staged into LDS with the
// async copy engine (global_load_async_to_lds_b32 + s_wait_asynccnt).
// Per face: neigh[a] += v[b]+v[c]; neigh[b] += v[a]+v[c]; neigh[c] += v[a]+v[b]
// ---------------------------------------------------------------------------
__global__ void __launch_bounds__(BLK)
scatter_kernel(const float* __restrict__ vert,
               const int* __restrict__ faces,
               float* __restrict__ neigh,
               int N, int F) {
    __shared__ int s_faces[BLK * 3];

    const int t = threadIdx.x;
    __builtin_assume(t < BLK);          // straight-line staging, full EXEC
    const int fbase = blockIdx.x * BLK;
    const int b     = blockIdx.y;

    // Stage BLK face triples (BLK*3 dwords) into LDS, coalesced across lanes.
    const int* gsrc   = faces + (size_t)fbase * 3;
    const int  navail = F * 3 - fbase * 3;  // dwords valid in this block
#if USE_ASYNC_LDS
    #pragma unroll
    for (int k = t; k < BLK * 3; k += BLK) {
        const int kk = (k < navail) ? k : 0;  // clamp OOB lanes to a safe addr
        __builtin_amdgcn_global_load_async_to_lds_b32(
            (global_int*)(gsrc + kk),
            (lds_int*)(s_faces + k),
            0, 0);
    }
    async_wait_all();
#else
    #pragma unroll
    for (int k = t; k < BLK * 3; k += BLK) {
        const int kk = (k < navail) ? k : 0;
        s_faces[k] = gsrc[kk];
    }
#endif
    __syncthreads();

    const int f = fbase + t;
    if (f >= F) return;

    const int ia = s_faces[3 * t + 0];
    const int ib = s_faces[3 * t + 1];
    const int ic = s_faces[3 * t + 2];

    const size_t N3 = (size_t)N * 3;
    const float* vb = vert + (size_t)b * N3;

    const float ax = vb[3 * (size_t)ia + 0];
    const float ay = vb[3 * (size_t)ia + 1];
    const float az = vb[3 * (size_t)ia + 2];
    const float bx = vb[3 * (size_t)ib + 0];
    const float by = vb[3 * (size_t)ib + 1];
    const float bz = vb[3 * (size_t)ib + 2];
    const float cx = vb[3 * (size_t)ic + 0];
    const float cy = vb[3 * (size_t)ic + 1];
    const float cz = vb[3 * (size_t)ic + 2];

    float* nb = neigh + (size_t)b * N3;
    // dst a receives src b and c
    atomAddF32(&nb[3 * (size_t)ia + 0], bx + cx);
    atomAddF32(&nb[3 * (size_t)ia + 1], by + cy);
    atomAddF32(&nb[3 * (size_t)ia + 2], bz + cz);
    // dst b receives src a and c
    atomAddF32(&nb[3 * (size_t)ib + 0], ax + cx);
    atomAddF32(&nb[3 * (size_t)ib + 1], ay + cy);
    atomAddF32(&nb[3 * (size_t)ib + 2], az + cz);
    // dst c receives src a and b
    atomAddF32(&nb[3 * (size_t)ic + 0], ax + bx);
    atomAddF32(&nb[3 * (size_t)ic + 1], ay + by);
    atomAddF32(&nb[3 * (size_t)ic + 2], az + bz);
}

// ---------------------------------------------------------------------------
// Kernel B: vertex degree. Each face contributes 2 incoming edges per vertex.
// ---------------------------------------------------------------------------
__global__ void __launch_bounds__(BLK)
deg_kernel(const int* __restrict__ faces, int* __restrict__ deg, int F) {
    int f = blockIdx.x * blockDim.x + threadIdx.x;
    if (f >= F) return;
    int a = faces[3 * f + 0];
    int b = faces[3 * f + 1];
    int c = faces[3 * f + 2];
    atomicAdd(&deg[a], 2);
    atomicAdd(&deg[b], 2);
    atomicAdd(&deg[c], 2);
}

// ---------------------------------------------------------------------------
// Kernel C: finalize. i = b*N + n.  lap = neigh/max(deg,1) - vert; out = |lap|
// ---------------------------------------------------------------------------
__global__ void __launch_bounds__(BLK)
finalize_kernel(const float* __restrict__ vert,
                const float* __restrict__ neigh,
                const int* __restrict__ deg,
                float* __restrict__ out,
                int N, int total) {
    int i = blockIdx.x * blockDim.x + threadIdx.x;
    if (i >= total) return;
    const int n = i % N;

    float d = (float)deg[n];
    d = fmaxf(d, 1.0f);

    const size_t base = (size_t)i * 3;
    const float lx = neigh[base + 0] / d - vert[base + 0];
    const float ly = neigh[base + 1] / d - vert[base + 1];
    const float lz = neigh[base + 2] / d - vert[base + 2];

    out[i] = sqrtf(lx * lx + ly * ly + lz * lz);
}

// ---------------------------------------------------------------------------
// Host launcher
// ---------------------------------------------------------------------------
extern "C" void kernel_launch(void* const* d_in, const int* in_sizes, int n_in,
                              void* d_out, int out_size, void* d_ws, size_t ws_size,
                              hipStream_t stream) {
    const float* vert  = (const float*)d_in[0];   // (B, N, 3) f32
    const int*   faces = (const int*)d_in[1];     // (F, 3) i32
    float*       out   = (float*)d_out;           // (B, N) f32

    const int F     = in_sizes[1] / 3;
    const int total = out_size;                   // B * N
    const int N     = total / BATCH;

    // Workspace: [neigh accumulator: total*3 f32][deg: N i32]
    float* neigh = (float*)d_ws;
    const size_t neigh_bytes = (size_t)total * 3 * sizeof(float);
    int* deg = (int*)((char*)d_ws + neigh_bytes);
    const size_t zero_bytes = neigh_bytes + (size_t)N * sizeof(int);

    (void)hipMemsetAsync(d_ws, 0, zero_bytes, stream);

    deg_kernel<<<(F + BLK - 1) / BLK, BLK, 0, stream>>>(faces, deg, F);

    dim3 grid((F + BLK - 1) / BLK, BATCH);
    scatter_kernel<<<grid, BLK, 0, stream>>>(vert, faces, neigh, N, F);

    finalize_kernel<<<(total + BLK - 1) / BLK, BLK, 0, stream>>>(
        vert, neigh, deg, out, N, total);
}